// N3Tree_23691039605429
// MI455X (gfx1250) — compile-verified
//
#include <hip/hip_runtime.h>
#include <hip/hip_bf16.h>

// Octree point query (svox N3Tree, N=2), MI455X / gfx1250.
//
//  - Stage the hot top of the child table (levels 0..3: 585 nodes * 8 ints =
//    18.3 KB) into LDS once per block via the CDNA5 async global->LDS path
//    (global_load_async_to_lds_b128 / ASYNCcnt / s_wait_asynccnt).
//  - Traversal reads the staged region with explicit ds_load_b32 inline asm
//    (compiler-proof: it cannot fold the asm-written LDS to undef or demote
//    the access off the LDS pipe). Deeper levels (128 KB + 1 MB) stay in the
//    192 MB L2.
//  - Grid-stride: ~8 queries/thread amortizes the staging.

#define BLOCK 256
#define LDS_NODES 585              // offs[4] = (8^4-1)/7 : levels 0..3 of a complete tree
#define LDS_INTS (LDS_NODES * 8)   // 4680 ints = 18720 bytes

struct alignas(4) Coord3 { float x, y, z; };   // 12B query coordinate record

// Raw LDS read at a workgroup-relative byte address. s_wait_dscnt is baked in
// because the compiler does not track counters for asm-issued DS ops.
__device__ __forceinline__ int lds_read_i32(unsigned byte_addr) {
    int v;
    asm volatile("ds_load_b32 %0, %1\n\t"
                 "s_wait_dscnt 0"
                 : "=v"(v) : "v"(byte_addr));
    return v;
}

__global__ __launch_bounds__(BLOCK) void octree_query_kernel(
    const float* __restrict__ indices,   // (Q,3)
    const float* __restrict__ leafdata,  // (n_leaves,4)
    const int*   __restrict__ child,     // (n_internal,8) flattened
    const float* __restrict__ scaling,   // (3,)
    const float* __restrict__ offs,      // (3,)
    const int*   __restrict__ depth_ptr, // scalar
    float*       __restrict__ out,       // (Q,4)
    int Q, int n_internal)
{
    __shared__ int lchild[LDS_INTS];

    // ---- Stage top-of-tree child rows into LDS via CDNA5 async copies ----
    const int stage_ints  = min(n_internal * 8, LDS_INTS);
    const int stage_nodes = stage_ints >> 3;
    // low 32 bits of a flat shared-aperture address == workgroup-relative LDS byte offset
    const unsigned lds_base = (unsigned)(uintptr_t)(&lchild[0]);

    const int nvec = stage_ints >> 2;             // 16-byte chunks
    for (int v = threadIdx.x; v < nvec; v += BLOCK) {
        unsigned la = lds_base + (unsigned)v * 16u;
        const int* gp = child + (size_t)v * 4;
        asm volatile("global_load_async_to_lds_b128 %0, %1, off"
                     :: "v"(la), "v"(gp) : "memory");
    }
    for (int v = (nvec << 2) + threadIdx.x; v < stage_ints; v += BLOCK) {
        unsigned la = lds_base + (unsigned)v * 4u;
        const int* gp = child + v;
        asm volatile("global_load_async_to_lds_b32 %0, %1, off"
                     :: "v"(la), "v"(gp) : "memory");
    }
    __builtin_amdgcn_s_wait_asynccnt(0);
    __syncthreads();

    // ---- Uniform params (scalar loads) ----
    const float sx = scaling[0], sy = scaling[1], sz = scaling[2];
    const float ox = offs[0],    oy = offs[1],    oz = offs[2];
    const int   depth = depth_ptr[0];
    const float CLAMP_HI = 1.0f - 1e-6f;

    const int tid    = blockIdx.x * BLOCK + threadIdx.x;
    const int stride = gridDim.x * BLOCK;

    for (int q = tid; q < Q; q += stride) {
        // world -> tree coords, clamp to [0, 1-1e-6]
        const Coord3 c = ((const Coord3*)indices)[q];
        float x = fminf(fmaxf(fmaf(c.x, sx, ox), 0.0f), CLAMP_HI);
        float y = fminf(fmaxf(fmaf(c.y, sy, oy), 0.0f), CLAMP_HI);
        float z = fminf(fmaxf(fmaf(c.z, sz, oz), 0.0f), CLAMP_HI);

        int  node = 0;
        int  didx = 0;
        bool done = false;

        for (int l = 0; l < depth; ++l) {
            // extract next octant bit per axis (exact in binary fp)
            x *= 2.0f; float fx = fminf(floorf(x), 1.0f); x -= fx;
            y *= 2.0f; float fy = fminf(floorf(y), 1.0f); y -= fy;
            z *= 2.0f; float fz = fminf(floorf(z), 1.0f); z -= fz;

            if (!done) {
                const int cidx = ((int)fx << 2) | ((int)fy << 1) | (int)fz;
                const int addr = node * 8 + cidx;
                const int delta = (node < stage_nodes)
                    ? lds_read_i32(lds_base + (unsigned)addr * 4u)   // ds_load_b32
                    : child[addr];                                   // L2-resident
                if (delta <= 0) {           // leaf: record data index, terminate
                    didx = -delta;
                    done = true;
                } else {                    // internal: descend
                    node += delta;
                }
            }
        }

        // leaf gather + coalesced 16B store
        const float4 v = ((const float4*)leafdata)[didx];
        ((float4*)out)[q] = v;
    }
}

extern "C" void kernel_launch(void* const* d_in, const int* in_sizes, int n_in,
                              void* d_out, int out_size, void* d_ws, size_t ws_size,
                              hipStream_t stream) {
    const float* indices = (const float*)d_in[0];
    const float* data    = (const float*)d_in[1];
    const int*   child   = (const int*)d_in[2];
    const float* scaling = (const float*)d_in[3];
    const float* offset  = (const float*)d_in[4];
    const int*   depth   = (const int*)d_in[5];
    float* out = (float*)d_out;

    const int Q          = in_sizes[0] / 3;
    const int n_internal = in_sizes[2] / 8;

    int blocks = (Q + BLOCK - 1) / BLOCK;
    if (blocks > 4096) blocks = 4096;   // ~8 queries/thread, amortizes LDS staging
    if (blocks < 1) blocks = 1;

    octree_query_kernel<<<blocks, BLOCK, 0, stream>>>(
        indices, data, child, scaling, offset, depth, out, Q, n_internal);
}